// GraphResNet_80925773791932
// MI455X (gfx1250) — compile-verified
//
#include <hip/hip_runtime.h>
#include <hip/hip_bf16.h>

// ---------------- problem constants (match reference) ----------------
#define NN      8192        // total nodes (B*NPER)
#define EE      49152       // total edges
#define NPER    4096
#define BATCH   2
#define NCC     450         // coarse clusters (B * 15 * 15)
#define NV5     225
#define NKEYS   (NCC * NCC) // coarse-edge key space
#define KTOT    512         // 8^3 spline kernel entries
#define KCHUNK  128         // k-chunk for coarse GEMM precompute
#define MTILES  29          // ceil(450/16)
#define NSLOT   32          // B * 16 final pool slots
#define CH7     128
#define NOUT    11
#define LDSW    132         // padded A-tile row stride (dwords) -> bank-conflict-free

typedef __attribute__((ext_vector_type(2))) float v2f;
typedef __attribute__((ext_vector_type(8))) float v8f;

// ---------------- small device helpers ----------------
__device__ __forceinline__ int iclamp(int v, int lo, int hi) {
  return v < lo ? lo : (v > hi ? hi : v);
}
__device__ __forceinline__ float eluf(float x) {
  return x > 0.0f ? x : (expf(x) - 1.0f);
}
// order-preserving float<->uint mapping for atomic max on floats
__device__ __forceinline__ unsigned int fenc(float x) {
  unsigned int u = __float_as_uint(x);
  return (u & 0x80000000u) ? ~u : (u | 0x80000000u);
}
__device__ __forceinline__ float fdec(unsigned int u) {
  return __uint_as_float((u & 0x80000000u) ? (u & 0x7FFFFFFFu) : ~u);
}

// ---------------- generic utility kernels ----------------
__global__ __launch_bounds__(256) void fill_enc_kernel(unsigned int* p, float v, int n) {
  int i = blockIdx.x * blockDim.x + threadIdx.x;
  if (i < n) p[i] = fenc(v);
}
__global__ __launch_bounds__(256) void add_kernel(const float* a, const float* b, float* o, int n) {
  int i = blockIdx.x * blockDim.x + threadIdx.x;
  if (i < n) o[i] = a[i] + b[i];
}

// ---------------- spline basis (degree-1 open B-spline, K=8, D=3) ----------------
__global__ __launch_bounds__(256) void spline_basis_kernel(const float* __restrict__ eattr,
                                                           float* __restrict__ bas,
                                                           int* __restrict__ wix, int nE) {
  int e = blockIdx.x * blockDim.x + threadIdx.x;
  if (e >= nE) return;
  float fr[3]; int lo[3];
#pragma unroll
  for (int d = 0; d < 3; ++d) {
    float p = eattr[e * 3 + d] * 7.0f;
    float fl = floorf(p);
    fr[d] = p - fl;
    lo[d] = (int)fl;
  }
#pragma unroll
  for (int s = 0; s < 8; ++s) {
    float b = 1.0f; int idx = 0; int pw = 1;
#pragma unroll
    for (int d = 0; d < 3; ++d) {
      int bit = (s >> d) & 1;
      b *= bit ? fr[d] : (1.0f - fr[d]);
      idx += iclamp(lo[d] + bit, 0, 7) * pw;
      pw *= 8;
    }
    bas[e * 8 + s] = b;
    wix[e * 8 + s] = idx;
  }
}

__global__ __launch_bounds__(256) void deg_kernel(const int* __restrict__ tgt, float* __restrict__ deg, int nE) {
  int e = blockIdx.x * blockDim.x + threadIdx.x;
  if (e < nE) atomicAdd(&deg[tgt[e]], 1.0f);
}

// ---------------- fine-graph spline conv: wave-per-edge, lane-per-out-channel ----
// msg[e] = sum_b bas[e,b] * x[src[e]] @ W[wix[e,b]] ; scatter-add into num[tgt].
// W tables are L2-resident; reads across the 32 lanes are 128B-coalesced.
template <int CIN, int COUT>
__global__ __launch_bounds__(256) void econv_kernel(const float* __restrict__ hin,
                                                    const float* __restrict__ W,
                                                    const float* __restrict__ bas,
                                                    const int* __restrict__ wix,
                                                    const int* __restrict__ src,
                                                    const int* __restrict__ tgt,
                                                    float* __restrict__ num, int nE) {
  int gid = blockIdx.x * blockDim.x + threadIdx.x;
  int e = gid >> 5;
  int lane = gid & 31;
  if (e >= nE) return;
  int s = src[e], t = tgt[e];
  float xj[CIN];
#pragma unroll
  for (int c = 0; c < CIN; ++c) xj[c] = hin[s * CIN + c];
  constexpr int NF = (COUT + 31) / 32;
  float acc[NF];
#pragma unroll
  for (int i = 0; i < NF; ++i) acc[i] = 0.0f;
  for (int b = 0; b < 8; ++b) {
    float bb = bas[e * 8 + b];
    int k = wix[e * 8 + b];
    const float* Wk = W + (size_t)k * (CIN * COUT);
    for (int fo = 0; fo < NF; ++fo) {
      int f = fo * 32 + lane;
      if (f < COUT) {
        float sd = 0.0f;
#pragma unroll
        for (int c = 0; c < CIN; ++c) sd = fmaf(xj[c], Wk[c * COUT + f], sd);
        acc[fo] = fmaf(bb, sd, acc[fo]);
      }
    }
  }
  for (int fo = 0; fo < NF; ++fo) {
    int f = fo * 32 + lane;
    if (f < COUT) atomicAdd(&num[(size_t)t * COUT + f], acc[fo]);
  }
}

// num/deg -> mean -> ELU (in place into out)
__global__ __launch_bounds__(256) void agg_elu_kernel(const float* __restrict__ num,
                                                      const float* __restrict__ deg,
                                                      float* __restrict__ out, int nRows, int C) {
  int i = blockIdx.x * blockDim.x + threadIdx.x;
  if (i >= nRows * C) return;
  int r = i / C;
  float d = fmaxf(deg[r], 1.0f);
  out[i] = eluf(num[i] / d);
}

// per-channel sum & sumsq (optionally masked); one block per channel
__global__ __launch_bounds__(256) void stats_kernel(const float* __restrict__ act, int nRows, int C,
                                                    const float* __restrict__ mask,
                                                    float* __restrict__ stats) {
  int f = blockIdx.x;
  int tid = threadIdx.x;
  float s = 0.0f, ss = 0.0f;
  for (int r = tid; r < nRows; r += blockDim.x) {
    float v = act[(size_t)r * C + f];
    float m = mask ? mask[r] : 1.0f;
    s += v * m;
    ss += v * v * m;
  }
  __shared__ float sh[512];
  sh[tid] = s; sh[256 + tid] = ss;
  __syncthreads();
  for (int off = 128; off > 0; off >>= 1) {
    if (tid < off) { sh[tid] += sh[tid + off]; sh[256 + tid] += sh[256 + tid + off]; }
    __syncthreads();
  }
  if (tid == 0) { stats[f] = sh[0]; stats[C + f] = sh[256]; }
}

__global__ __launch_bounds__(256) void bn_kernel(const float* __restrict__ act, float* __restrict__ out,
                                                 const float* __restrict__ stats,
                                                 const float* __restrict__ g, const float* __restrict__ b,
                                                 int nRows, int C, const float* __restrict__ cntptr) {
  int i = blockIdx.x * blockDim.x + threadIdx.x;
  if (i >= nRows * C) return;
  int f = i % C;
  float cnt = cntptr ? fmaxf(cntptr[0], 1.0f) : (float)nRows;
  float m = stats[f] / cnt;
  float v = stats[C + f] / cnt - m * m;
  out[i] = (act[i] - m) * rsqrtf(v + 1e-5f) * g[f] + b[f];
}

// ---------------- voxel pooling (16x12 grid, 15x15 per sample) ----------------
__global__ __launch_bounds__(256) void cluster_kernel(const float* __restrict__ pos,
                                                      const int* __restrict__ batch,
                                                      int* __restrict__ cl, float* __restrict__ cnt,
                                                      float* __restrict__ pospsum) {
  int n = blockIdx.x * blockDim.x + threadIdx.x;
  if (n >= NN) return;
  float px = pos[n * 3 + 0], py = pos[n * 3 + 1];
  int vx = iclamp((int)floorf(px / 16.0f), 0, 14);
  int vy = iclamp((int)floorf(py / 12.0f), 0, 14);
  int c = batch[n] * NV5 + vy * 15 + vx;
  cl[n] = c;
  atomicAdd(&cnt[c], 1.0f);
  for (int d = 0; d < 3; ++d) atomicAdd(&pospsum[c * 3 + d], pos[n * 3 + d]);
}

__global__ __launch_bounds__(256) void posp_valid_kernel(const float* __restrict__ cnt,
                                                         const float* __restrict__ pospsum,
                                                         float* __restrict__ posp,
                                                         float* __restrict__ validf,
                                                         float* __restrict__ validcnt) {
  int c = blockIdx.x * blockDim.x + threadIdx.x;
  if (c >= NCC) return;
  float ct = cnt[c];
  float vf = ct > 0.0f ? 1.0f : 0.0f;
  validf[c] = vf;
  atomicAdd(validcnt, vf);
  float inv = 1.0f / fmaxf(ct, 1.0f);
  for (int d = 0; d < 3; ++d) posp[c * 3 + d] = pospsum[c * 3 + d] * inv;
}

__global__ __launch_bounds__(256) void xpmax_kernel(const float* __restrict__ h,
                                                    const int* __restrict__ cl,
                                                    unsigned int* __restrict__ xpbits) {
  int i = blockIdx.x * blockDim.x + threadIdx.x;
  if (i >= NN * 128) return;
  int n = i >> 7, f = i & 127;
  atomicMax(&xpbits[(size_t)cl[n] * 128 + f], fenc(h[i]));
}

__global__ __launch_bounds__(256) void xp_finalize_kernel(const unsigned int* __restrict__ xpbits,
                                                          const float* __restrict__ cnt,
                                                          float* __restrict__ xp) {
  int i = blockIdx.x * blockDim.x + threadIdx.x;
  if (i >= NCC * 128) return;
  int c = i >> 7;
  xp[i] = cnt[c] > 0.0f ? fdec(xpbits[i]) : 0.0f;
}

// ---------------- coarse edge construction (dedupe via key bitmap) ----------------
__global__ __launch_bounds__(256) void mark_present_kernel(const int* __restrict__ src,
                                                           const int* __restrict__ tgt,
                                                           const int* __restrict__ cl,
                                                           unsigned char* __restrict__ present, int nE) {
  int e = blockIdx.x * blockDim.x + threadIdx.x;
  if (e < nE) present[(size_t)cl[src[e]] * NCC + cl[tgt[e]]] = 1;
}

__global__ __launch_bounds__(256) void coarse_prep_kernel(const unsigned char* __restrict__ present,
                                                          const float* __restrict__ posp,
                                                          unsigned int* __restrict__ amaxbits,
                                                          float* __restrict__ cdeg) {
  int key = blockIdx.x * blockDim.x + threadIdx.x;
  if (key >= NKEYS) return;
  if (!present[key]) return;
  int s = key / NCC, t = key - s * NCC;
  if (s == t) return;
  float a = 0.0f;
  for (int d = 0; d < 3; ++d) a = fmaxf(a, fabsf(posp[s * 3 + d] - posp[t * 3 + d]));
  atomicMax(amaxbits, __float_as_uint(a));  // non-negative floats: bits are order-preserving
  atomicAdd(&cdeg[t], 1.0f);
}

// ---------------- coarse precompute GEMM: Y[k] = X(450x128) @ W[k](128x128) ---------
// One block per (k, mtile): the 16x128 A-tile is staged (zero-padded) into LDS once
// and shared by all 8 waves; each wave produces one 16x16 tile via 32x
// V_WMMA_F32_16X16X4_F32 in a branch-free loop (ds_load A, global_load B, prefetch W).
__global__ __launch_bounds__(256) void gemm_wmma_kernel(const float* __restrict__ X,
                                                        const float* __restrict__ W,
                                                        float* __restrict__ Y, int kcount) {
  __shared__ float Atile[16 * LDSW];
  int mt = blockIdx.x % MTILES;
  int k = blockIdx.x / MTILES;          // uniform per block
  if (k >= kcount) return;
  // cooperative, branch-light staging of the A tile (rows >= NCC -> 0)
  for (int i = threadIdx.x; i < 16 * 128; i += 256) {
    int r = i >> 7, c = i & 127;
    int row = mt * 16 + r;
    Atile[r * LDSW + c] = (row < NCC) ? X[(size_t)row * 128 + c] : 0.0f;
  }
  __syncthreads();

  int lane = threadIdx.x & 31;
  int nt = threadIdx.x >> 5;            // wave id = N-tile
  const float* Wk = W + (size_t)k * (128 * 128);
  int ncol = nt * 16 + (lane & 15);
  int koff = (lane >> 4) * 2;           // ISA A/B layout: lanes 16-31 hold K+2,K+3
  int arow = lane & 15;
  v8f acc = {0.f, 0.f, 0.f, 0.f, 0.f, 0.f, 0.f, 0.f};
  for (int kk = 0; kk < 128; kk += 4) {
    v2f a, b;
    a.x = Atile[arow * LDSW + kk + koff];
    a.y = Atile[arow * LDSW + kk + koff + 1];
    b.x = Wk[(size_t)(kk + koff) * 128 + ncol];
    b.y = Wk[(size_t)(kk + koff + 1) * 128 + ncol];
    if (kk + 8 < 128) __builtin_prefetch(&Wk[(size_t)(kk + 8) * 128 + ncol], 0, 3);
    acc = __builtin_amdgcn_wmma_f32_16x16x4_f32(false, a, false, b, (short)0, acc, false, false);
  }
  float* Yk = Y + (size_t)k * (NCC * 128);
  int mbase = mt * 16 + ((lane >> 4) * 8);  // C/D layout: VGPR j -> M=j (+8 for hi lanes)
  for (int j = 0; j < 8; ++j) {
    int m = mbase + j;
    if (m < NCC) Yk[(size_t)m * 128 + ncol] = acc[j];
  }
}

// ---------------- coarse edge message accumulate (wave per unique key) -------------
__global__ __launch_bounds__(256) void coarse_accum_kernel(const float* __restrict__ Y,
                                                           const float* __restrict__ posp,
                                                           const unsigned char* __restrict__ present,
                                                           const unsigned int* __restrict__ amaxbits,
                                                           float* __restrict__ num2, int kc0) {
  int gid = blockIdx.x * blockDim.x + threadIdx.x;
  int key = gid >> 5;
  int lane = gid & 31;
  if (key >= NKEYS) return;
  if (!present[key]) return;
  int s = key / NCC, t = key - s * NCC;
  if (s == t) return;
  float denom = 2.0f * __uint_as_float(amaxbits[0]) + 1e-12f;
  float fr[3]; int lo[3];
  for (int d = 0; d < 3; ++d) {
    float cart = posp[s * 3 + d] - posp[t * 3 + d];
    float ps = fminf(fmaxf(cart / denom + 0.5f, 0.0f), 1.0f);
    float p = ps * 7.0f;
    float fl = floorf(p);
    fr[d] = p - fl;
    lo[d] = (int)fl;
  }
  for (int b = 0; b < 8; ++b) {
    float bb = 1.0f; int idx = 0; int pw = 1;
    for (int d = 0; d < 3; ++d) {
      int bit = (b >> d) & 1;
      bb *= bit ? fr[d] : (1.0f - fr[d]);
      idx += iclamp(lo[d] + bit, 0, 7) * pw;
      pw *= 8;
    }
    if (idx >= kc0 && idx < kc0 + KCHUNK) {
      const float* Yrow = Y + ((size_t)(idx - kc0) * NCC + s) * 128;
      for (int f = lane; f < 128; f += 32)
        atomicAdd(&num2[(size_t)t * 128 + f], bb * Yrow[f]);
    }
  }
}

// ---------------- final 4x4 pool + FC ----------------
__global__ __launch_bounds__(256) void pool7_kernel(const float* __restrict__ h2,
                                                    const float* __restrict__ posp,
                                                    const float* __restrict__ validf,
                                                    unsigned int* __restrict__ x7bits,
                                                    float* __restrict__ cnt7) {
  int i = blockIdx.x * blockDim.x + threadIdx.x;
  if (i >= NCC * 128) return;
  int c = i >> 7, f = i & 127;
  float px = posp[c * 3 + 0], py = posp[c * 3 + 1];
  int vx = iclamp((int)floorf(px / 60.0f), 0, 3);
  int vy = iclamp((int)floorf(py / 45.0f), 0, 3);
  int slot = (c / NV5) * 16 + vy * 4 + vx;
  float hv = validf[c] > 0.0f ? h2[i] : -1e30f;
  atomicMax(&x7bits[(size_t)slot * 128 + f], fenc(hv));
  if (f == 0) atomicAdd(&cnt7[slot], validf[c]);
}

__global__ __launch_bounds__(256) void x7_finalize_kernel(const unsigned int* __restrict__ x7bits,
                                                          const float* __restrict__ cnt7,
                                                          float* __restrict__ x7) {
  int i = blockIdx.x * blockDim.x + threadIdx.x;
  if (i >= NSLOT * 128) return;
  int slot = i >> 7;
  x7[i] = cnt7[slot] > 0.0f ? fdec(x7bits[i]) : 0.0f;
}

__global__ __launch_bounds__(256) void fc_kernel(const float* __restrict__ x7,
                                                 const float* __restrict__ Wfc,
                                                 float* __restrict__ out) {
  int bo = blockIdx.x;            // 0..21 = b*11+o
  int b_ = bo / NOUT, o = bo % NOUT;
  float partial = 0.0f;
  for (int j = threadIdx.x; j < 16 * CH7; j += blockDim.x)
    partial += x7[b_ * (16 * CH7) + j] * Wfc[(size_t)o * (16 * CH7) + j];
  __shared__ float sh[256];
  sh[threadIdx.x] = partial;
  __syncthreads();
  for (int off = 128; off > 0; off >>= 1) {
    if (threadIdx.x < off) sh[threadIdx.x] += sh[threadIdx.x + off];
    __syncthreads();
  }
  if (threadIdx.x == 0) out[bo] = sh[0];
}

// =====================================================================
extern "C" void kernel_launch(void* const* d_in, const int* in_sizes, int n_in,
                              void* d_out, int out_size, void* d_ws, size_t ws_size,
                              hipStream_t stream) {
  (void)in_sizes; (void)n_in; (void)out_size; (void)ws_size;
  const float* x      = (const float*)d_in[0];
  const float* pos    = (const float*)d_in[1];
  const float* eattr  = (const float*)d_in[2];
  const int*   eidx   = (const int*)d_in[3];
  const int*   batch  = (const int*)d_in[4];
  const int* src = eidx;
  const int* tgt = eidx + EE;
  const float* W[8]; const float* gg[8]; const float* bb[8];
  for (int i = 1; i <= 7; ++i) {
    W[i]  = (const float*)d_in[5 + 3 * (i - 1) + 0];
    gg[i] = (const float*)d_in[5 + 3 * (i - 1) + 1];
    bb[i] = (const float*)d_in[5 + 3 * (i - 1) + 2];
  }
  const float* Wfc = (const float*)d_in[26];
  float* out = (float*)d_out;

  // ---- bump allocator over workspace (~50 MB total) ----
  char* wp = (char*)d_ws;
  auto alloc = [&](size_t bytes) -> void* {
    void* r = (void*)wp;
    wp += (bytes + 255) & ~(size_t)255;
    return r;
  };
  float* bas      = (float*)alloc((size_t)EE * 8 * 4);
  int*   wix      = (int*)  alloc((size_t)EE * 8 * 4);
  float* deg      = (float*)alloc((size_t)NN * 4);
  float* hA       = (float*)alloc((size_t)NN * 128 * 4);
  float* hB       = (float*)alloc((size_t)NN * 128 * 4);
  float* hC       = (float*)alloc((size_t)NN * 128 * 4);
  float* num      = (float*)alloc((size_t)NN * 128 * 4);
  float* stats    = (float*)alloc(256 * 4);
  int*   cl       = (int*)  alloc((size_t)NN * 4);
  float* cnt      = (float*)alloc((size_t)NCC * 4);
  float* pospsum  = (float*)alloc((size_t)NCC * 3 * 4);
  float* posp     = (float*)alloc((size_t)NCC * 3 * 4);
  float* validf   = (float*)alloc((size_t)NCC * 4);
  float* validcnt = (float*)alloc(4);
  unsigned int* xpbits = (unsigned int*)alloc((size_t)NCC * 128 * 4);
  float* xp       = (float*)alloc((size_t)NCC * 128 * 4);
  unsigned char* present = (unsigned char*)alloc((size_t)NKEYS);
  unsigned int* amaxbits = (unsigned int*)alloc(4);
  float* cdeg     = (float*)alloc((size_t)NCC * 4);
  float* num2     = (float*)alloc((size_t)NCC * 128 * 4);
  float* hc1      = (float*)alloc((size_t)NCC * 128 * 4);
  float* hc2      = (float*)alloc((size_t)NCC * 128 * 4);
  float* Ychunk   = (float*)alloc((size_t)KCHUNK * NCC * 128 * 4);
  unsigned int* x7bits = (unsigned int*)alloc((size_t)NSLOT * 128 * 4);
  float* cnt7     = (float*)alloc((size_t)NSLOT * 4);
  float* x7       = (float*)alloc((size_t)NSLOT * 128 * 4);

  auto gridFor = [](long n) { return (int)((n + 255) / 256); };

  // ---- fine-graph basis + degrees ----
  spline_basis_kernel<<<gridFor(EE), 256, 0, stream>>>(eattr, bas, wix, EE);
  hipMemsetAsync(deg, 0, (size_t)NN * 4, stream);
  deg_kernel<<<gridFor(EE), 256, 0, stream>>>(tgt, deg, EE);

  const int econvBlocks = gridFor((long)EE * 32);

  // per fine layer: conv -> mean -> ELU -> BN (in place)
  auto fine_layer = [&](auto econvLauncher, const float* hin, float* hout,
                        int Cout, const float* g, const float* b) {
    hipMemsetAsync(num, 0, (size_t)NN * Cout * 4, stream);
    econvLauncher(hin);
    agg_elu_kernel<<<gridFor((long)NN * Cout), 256, 0, stream>>>(num, deg, hout, NN, Cout);
    stats_kernel<<<Cout, 256, 0, stream>>>(hout, NN, Cout, nullptr, stats);
    bn_kernel<<<gridFor((long)NN * Cout), 256, 0, stream>>>(hout, hout, stats, g, b, NN, Cout, nullptr);
  };

  // L1: x(1) -> hA(16)
  fine_layer([&](const float* hin) {
    econv_kernel<1, 16><<<econvBlocks, 256, 0, stream>>>(hin, W[1], bas, wix, src, tgt, num, EE);
  }, x, hA, 16, gg[1], bb[1]);
  // L2: hA(16) -> hB(32)   (hB doubles as residual "sc")
  fine_layer([&](const float* hin) {
    econv_kernel<16, 32><<<econvBlocks, 256, 0, stream>>>(hin, W[2], bas, wix, src, tgt, num, EE);
  }, hA, hB, 32, gg[2], bb[2]);
  // L3: hB(32) -> hA(32)
  fine_layer([&](const float* hin) {
    econv_kernel<32, 32><<<econvBlocks, 256, 0, stream>>>(hin, W[3], bas, wix, src, tgt, num, EE);
  }, hB, hA, 32, gg[3], bb[3]);
  // L4: hA(32) -> hC(32); then hA = hC + hB (residual)
  fine_layer([&](const float* hin) {
    econv_kernel<32, 32><<<econvBlocks, 256, 0, stream>>>(hin, W[4], bas, wix, src, tgt, num, EE);
  }, hA, hC, 32, gg[4], bb[4]);
  add_kernel<<<gridFor((long)NN * 32), 256, 0, stream>>>(hC, hB, hA, NN * 32);
  // L5: hA(32) -> hB(128)
  fine_layer([&](const float* hin) {
    econv_kernel<32, 128><<<econvBlocks, 256, 0, stream>>>(hin, W[5], bas, wix, src, tgt, num, EE);
  }, hA, hB, 128, gg[5], bb[5]);

  // ---- voxel pool (16x12) ----
  hipMemsetAsync(cnt, 0, (size_t)NCC * 4, stream);
  hipMemsetAsync(pospsum, 0, (size_t)NCC * 3 * 4, stream);
  hipMemsetAsync(validcnt, 0, 4, stream);
  cluster_kernel<<<gridFor(NN), 256, 0, stream>>>(pos, batch, cl, cnt, pospsum);
  posp_valid_kernel<<<gridFor(NCC), 256, 0, stream>>>(cnt, pospsum, posp, validf, validcnt);
  fill_enc_kernel<<<gridFor((long)NCC * 128), 256, 0, stream>>>(xpbits, -__builtin_huge_valf(), NCC * 128);
  xpmax_kernel<<<gridFor((long)NN * 128), 256, 0, stream>>>(hB, cl, xpbits);
  xp_finalize_kernel<<<gridFor((long)NCC * 128), 256, 0, stream>>>(xpbits, cnt, xp);

  // ---- coarse edge structure (bitmap dedupe == sort/uniq/self-loop-drop) ----
  hipMemsetAsync(present, 0, (size_t)NKEYS, stream);
  hipMemsetAsync(amaxbits, 0, 4, stream);
  hipMemsetAsync(cdeg, 0, (size_t)NCC * 4, stream);
  mark_present_kernel<<<gridFor(EE), 256, 0, stream>>>(src, tgt, cl, present, EE);
  coarse_prep_kernel<<<gridFor(NKEYS), 256, 0, stream>>>(present, posp, amaxbits, cdeg);

  // ---- coarse layers 6 & 7: WMMA precompute Y = X @ W[k], chunked in k ----
  const int gemmBlocks = KCHUNK * MTILES;     // one block per (k, mtile)
  const int accumBlocks = gridFor((long)NKEYS * 32);
  auto coarse_layer = [&](const float* hin, float* hout, const float* Wt,
                          const float* g, const float* b) {
    hipMemsetAsync(num2, 0, (size_t)NCC * 128 * 4, stream);
    for (int kc0 = 0; kc0 < KTOT; kc0 += KCHUNK) {
      gemm_wmma_kernel<<<gemmBlocks, 256, 0, stream>>>(hin, Wt + (size_t)kc0 * 128 * 128, Ychunk, KCHUNK);
      coarse_accum_kernel<<<accumBlocks, 256, 0, stream>>>(Ychunk, posp, present, amaxbits, num2, kc0);
    }
    agg_elu_kernel<<<gridFor((long)NCC * 128), 256, 0, stream>>>(num2, cdeg, hout, NCC, 128);
    stats_kernel<<<128, 256, 0, stream>>>(hout, NCC, 128, validf, stats);
    bn_kernel<<<gridFor((long)NCC * 128), 256, 0, stream>>>(hout, hout, stats, g, b, NCC, 128, validcnt);
  };
  coarse_layer(xp, hc1, W[6], gg[6], bb[6]);
  coarse_layer(hc1, hc2, W[7], gg[7], bb[7]);
  add_kernel<<<gridFor((long)NCC * 128), 256, 0, stream>>>(hc2, xp, hc2, NCC * 128);

  // ---- final 4x4 pool + FC ----
  fill_enc_kernel<<<gridFor((long)NSLOT * 128), 256, 0, stream>>>(x7bits, -__builtin_huge_valf(), NSLOT * 128);
  hipMemsetAsync(cnt7, 0, (size_t)NSLOT * 4, stream);
  pool7_kernel<<<gridFor((long)NCC * 128), 256, 0, stream>>>(hc2, posp, validf, x7bits, cnt7);
  x7_finalize_kernel<<<gridFor((long)NSLOT * 128), 256, 0, stream>>>(x7bits, cnt7, x7);
  fc_kernel<<<BATCH * NOUT, 256, 0, stream>>>(x7, Wfc, out);
}